// Pure_3DMM_18287970746897
// MI455X (gfx1250) — compile-verified
//
#include <hip/hip_runtime.h>
#include <hip/hip_bf16.h>
#include <math.h>

typedef float v2f __attribute__((ext_vector_type(2)));
typedef float v8f __attribute__((ext_vector_type(8)));

#define FOCALF 1015.0f
#define HALF_IMG 128.0f
#define IMGF 256.0f
#define B_ROWS 128
#define CSTRIDE 257
#define LDA_PAD 82   // 128 rows x 82 floats: conflict-free (18m+k mod 64 distinct), 8B-aligned even k

// ---------------------------------------------------------------------------
// Kernel A: per-batch precompute — rotation M = Rz*Ry*Rx (row-major M[d][c]),
// gamma_adj[b][9][3] (gamma+0.8 on band 0, transposed), meanshape centroid.
// ---------------------------------------------------------------------------
__global__ __launch_bounds__(128) void precompute_kernel(
    const float* __restrict__ coeff, const float* __restrict__ meanshape,
    int NV, float* __restrict__ rot, float* __restrict__ gamma_adj,
    float* __restrict__ center)
{
    __shared__ float part[128 * 3];
    const int tid = threadIdx.x;

    float s0 = 0.f, s1 = 0.f, s2 = 0.f;
    for (int v = tid; v < NV; v += 128) {
        s0 += meanshape[v * 3 + 0];
        s1 += meanshape[v * 3 + 1];
        s2 += meanshape[v * 3 + 2];
    }
    part[tid * 3 + 0] = s0; part[tid * 3 + 1] = s1; part[tid * 3 + 2] = s2;
    __syncthreads();
    if (tid < 3) {
        float acc = 0.f;
        for (int t = 0; t < 128; ++t) acc += part[t * 3 + tid];
        center[tid] = acc / (float)NV;
    }

    if (tid < B_ROWS) {
        const float* c = coeff + (size_t)tid * CSTRIDE;
        float sx = sinf(c[224]), cx = cosf(c[224]);
        float sy = sinf(c[225]), cy = cosf(c[225]);
        float sz = sinf(c[226]), cz = cosf(c[226]);
        float* M = rot + tid * 9;
        // M = Rz * Ry * Rx
        M[0] = cz * cy;  M[1] = cz * sy * sx - sz * cx;  M[2] = cz * sy * cx + sz * sx;
        M[3] = sz * cy;  M[4] = sz * sy * sx + cz * cx;  M[5] = sz * sy * cx - cz * sx;
        M[6] = -sy;      M[7] = cy * sx;                 M[8] = cy * cx;

        float* G = gamma_adj + tid * 27;
        for (int j = 0; j < 9; ++j)
            for (int ch = 0; ch < 3; ++ch)
                G[j * 3 + ch] = c[227 + ch * 9 + j] + (j == 0 ? 0.8f : 0.0f);
    }
}

// ---------------------------------------------------------------------------
// WMMA GEMM: out[b, n] = sum_k coeff[b, acol+k] * base[n, k] + bias[n] (- center[n%3])
// M=128 (8 tiles of 16), one wave per 16-column strip, V_WMMA_F32_16X16X4_F32.
// A layout (16x4): lane l holds m=l&15, k = 2*(l>>4)+j (j = vector elem).
// B layout (4x16): lane l holds n=l&15, k = 2*(l>>4)+j.
// C layout (16x16): vgpr r, lane l -> m = r + 8*(l>>4), n = l&15.
// Inner loop: explicit load-phase (2 global b64 + 16 ds b64) then 16 WMMAs
// back-to-back so the scheduler overlaps next loads with the WMMA stream.
// ---------------------------------------------------------------------------
__global__ __launch_bounds__(256) void gemm16_wmma(
    const float* __restrict__ coeff, int acol0,
    const float* __restrict__ base0, int K0,
    const float* __restrict__ base1, int K1,
    const float* __restrict__ bias, const float* __restrict__ center,
    float* __restrict__ out, int NV3)
{
    __shared__ float ldsA[128 * LDA_PAD];
    const int tid   = threadIdx.x;
    const int lane  = tid & 31;
    const int wave  = tid >> 5;
    const int ncol  = lane & 15;
    const int khalf = lane >> 4;
    const int n0    = (blockIdx.x * 8 + wave) * 16;
    const int n     = n0 + ncol;
    const int nclmp = (n < NV3) ? n : (NV3 - 1);

    v8f acc[8];
#pragma unroll
    for (int mt = 0; mt < 8; ++mt) acc[mt] = (v8f){0.f,0.f,0.f,0.f,0.f,0.f,0.f,0.f};

    for (int seg = 0; seg < 2; ++seg) {
        const int K = seg ? K1 : K0;
        if (K <= 0) continue;
        const float* base = seg ? base1 : base0;
        const int acol    = seg ? (acol0 + K0) : acol0;

        __syncthreads();  // protect LDS reuse across segments
        {   // stage coeff A-block: 2 threads per row, interleaved columns
            const int m = tid >> 1, h = tid & 1;
            for (int k = h; k < K; k += 2)
                ldsA[m * LDA_PAD + k] = coeff[(size_t)m * CSTRIDE + acol + k];
        }
        __syncthreads();

        const float* brow = base + (size_t)nclmp * K;
        const int K8 = K & ~7;

        for (int kb = 0; kb < K8; kb += 8) {
            const int kk = kb + 2 * khalf;
            __builtin_prefetch(brow + kb + 64, 0, 1);

            // ---- load phase: 2 global b64 + 16 ds b64 into distinct regs ----
            v2f b0, b1;
            b0.x = brow[kk];     b0.y = brow[kk + 1];
            b1.x = brow[kk + 4]; b1.y = brow[kk + 5];
            v2f a0[8], a1[8];
#pragma unroll
            for (int mt = 0; mt < 8; ++mt) {
                const float* lp = &ldsA[(mt * 16 + ncol) * LDA_PAD + kk];
                a0[mt].x = lp[0]; a0[mt].y = lp[1];
                a1[mt].x = lp[4]; a1[mt].y = lp[5];
            }
            // ---- compute phase: 16 WMMAs back-to-back ----
#pragma unroll
            for (int mt = 0; mt < 8; ++mt)
                acc[mt] = __builtin_amdgcn_wmma_f32_16x16x4_f32(
                    false, a0[mt], false, b0, (short)0, acc[mt], false, false);
#pragma unroll
            for (int mt = 0; mt < 8; ++mt)
                acc[mt] = __builtin_amdgcn_wmma_f32_16x16x4_f32(
                    false, a1[mt], false, b1, (short)0, acc[mt], false, false);
        }

        // K%8 tail (single k-step of 4)
        for (int kb = K8; kb < K; kb += 4) {
            const int kk = kb + 2 * khalf;
            v2f b0;
            b0.x = brow[kk]; b0.y = brow[kk + 1];
            v2f a0[8];
#pragma unroll
            for (int mt = 0; mt < 8; ++mt) {
                const float* lp = &ldsA[(mt * 16 + ncol) * LDA_PAD + kk];
                a0[mt].x = lp[0]; a0[mt].y = lp[1];
            }
#pragma unroll
            for (int mt = 0; mt < 8; ++mt)
                acc[mt] = __builtin_amdgcn_wmma_f32_16x16x4_f32(
                    false, a0[mt], false, b0, (short)0, acc[mt], false, false);
        }
    }

    if (n < NV3) {
        float bv = bias[n];
        if (center) bv -= center[n - (n / 3) * 3];
#pragma unroll
        for (int mt = 0; mt < 8; ++mt) {
#pragma unroll
            for (int r = 0; r < 8; ++r) {
                const int b = mt * 16 + 8 * khalf + r;
                out[(size_t)b * NV3 + n] = acc[mt][r] + bv;
            }
        }
    }
}

// ---------------------------------------------------------------------------
// Face normals: fn[b, f] = cross(v1-v2, v2-v3) from centered shape; row NF = 0.
// ---------------------------------------------------------------------------
__global__ __launch_bounds__(256) void face_normal_kernel(
    const float* __restrict__ shape, const int* __restrict__ tri,
    float* __restrict__ fn, int NV3, int NF)
{
    const long long idx = (long long)blockIdx.x * blockDim.x + threadIdx.x;
    const long long total = (long long)B_ROWS * (NF + 1);
    if (idx >= total) return;
    const int f = (int)(idx % (NF + 1));
    const int b = (int)(idx / (NF + 1));
    float r0 = 0.f, r1 = 0.f, r2 = 0.f;
    if (f < NF) {
        const int i0 = tri[f * 3 + 0] - 1;
        const int i1 = tri[f * 3 + 1] - 1;
        const int i2 = tri[f * 3 + 2] - 1;
        const float* sb = shape + (size_t)b * NV3;
        const float ax = sb[i0*3+0] - sb[i1*3+0];
        const float ay = sb[i0*3+1] - sb[i1*3+1];
        const float az = sb[i0*3+2] - sb[i1*3+2];
        const float bx = sb[i1*3+0] - sb[i2*3+0];
        const float by = sb[i1*3+1] - sb[i2*3+1];
        const float bz = sb[i1*3+2] - sb[i2*3+2];
        r0 = ay * bz - az * by;
        r1 = az * bx - ax * bz;
        r2 = ax * by - ay * bx;
    }
    float* o = fn + ((size_t)b * (NF + 1) + f) * 3;
    o[0] = r0; o[1] = r1; o[2] = r2;
}

// ---------------------------------------------------------------------------
// Rigid transform in place: shape_t = shape @ rot + trans (per-vertex RMW).
// ---------------------------------------------------------------------------
__global__ __launch_bounds__(256) void rigid_kernel(
    float* __restrict__ shape_io, const float* __restrict__ coeff,
    const float* __restrict__ rot, int NV)
{
    const long long idx = (long long)blockIdx.x * blockDim.x + threadIdx.x;
    if (idx >= (long long)B_ROWS * NV) return;
    const int v = (int)(idx % NV);
    const int b = (int)(idx / NV);
    float* p = shape_io + ((size_t)b * NV + v) * 3;
    const float s0 = p[0], s1 = p[1], s2 = p[2];
    const float* M = rot + b * 9;
    const float* tr = coeff + (size_t)b * CSTRIDE + 254;
    p[0] = M[0] * s0 + M[1] * s1 + M[2] * s2 + tr[0];
    p[1] = M[3] * s0 + M[4] * s1 + M[5] * s2 + tr[1];
    p[2] = M[6] * s0 + M[7] * s1 + M[8] * s2 + tr[2];
}

// ---------------------------------------------------------------------------
// Vertex normals (point_buf gather) + rotate + SH illumination; tex -> color in place.
// ---------------------------------------------------------------------------
__global__ __launch_bounds__(256) void shade_kernel(
    float* __restrict__ tex_io, const float* __restrict__ fn,
    const int* __restrict__ point_buf, const float* __restrict__ rot,
    const float* __restrict__ gamma_adj, int NV, int NF)
{
    const long long idx = (long long)blockIdx.x * blockDim.x + threadIdx.x;
    if (idx >= (long long)B_ROWS * NV) return;
    const int v = (int)(idx % NV);
    const int b = (int)(idx / NV);

    const float* fb = fn + (size_t)b * (NF + 1) * 3;
    float n0 = 0.f, n1 = 0.f, n2 = 0.f;
#pragma unroll
    for (int j = 0; j < 8; ++j) {
        const int f = point_buf[(size_t)v * 8 + j] - 1;
        const float* fp = fb + (size_t)f * 3;
        n0 += fp[0]; n1 += fp[1]; n2 += fp[2];
    }
    const float inv = 1.0f / sqrtf(n0 * n0 + n1 * n1 + n2 * n2);
    n0 *= inv; n1 *= inv; n2 *= inv;

    const float* M = rot + b * 9;
    const float nx = M[0] * n0 + M[1] * n1 + M[2] * n2;
    const float ny = M[3] * n0 + M[4] * n1 + M[5] * n2;
    const float nz = M[6] * n0 + M[7] * n1 + M[8] * n2;

    const float Y0 = 0.88622695f;
    const float Y1 = -1.77245390f * ny;
    const float Y2 =  1.77245390f * nz;
    const float Y3 = -1.77245390f * nx;
    const float Y4 =  2.42705917f * nx * ny;
    const float Y5 = -2.42705917f * ny * nz;
    const float Y6 =  0.70062926f * (3.f * nz * nz - 1.f);
    const float Y7 = -2.42705917f * nx * nz;
    const float Y8 =  1.21352958f * (nx * nx - ny * ny);

    const float* G = gamma_adj + b * 27;
    float* t = tex_io + ((size_t)b * NV + v) * 3;
#pragma unroll
    for (int d = 0; d < 3; ++d) {
        const float L = Y0 * G[0 + d] + Y1 * G[3 + d] + Y2 * G[6 + d] +
                        Y3 * G[9 + d] + Y4 * G[12 + d] + Y5 * G[15 + d] +
                        Y6 * G[18 + d] + Y7 * G[21 + d] + Y8 * G[24 + d];
        t[d] = t[d] * L;
    }
}

// ---------------------------------------------------------------------------
// Landmark projection of shape_t at kp_inds.
// ---------------------------------------------------------------------------
__global__ __launch_bounds__(256) void lms_kernel(
    const float* __restrict__ shape_t, const int* __restrict__ kp,
    float* __restrict__ lms, int NV, int NKP)
{
    const int idx = blockIdx.x * blockDim.x + threadIdx.x;
    if (idx >= B_ROWS * NKP) return;
    const int i = idx % NKP;
    const int b = idx / NKP;
    const int v = kp[i];
    const float* p = shape_t + ((size_t)b * NV + v) * 3;
    const float z = 10.0f - p[2];
    const float invz = 1.0f / z;
    const float x = (FOCALF * p[0] + HALF_IMG * z) * invz;
    const float y = (FOCALF * p[1] + HALF_IMG * z) * invz;
    lms[(size_t)idx * 2 + 0] = x;
    lms[(size_t)idx * 2 + 1] = IMGF - y;
}

// ---------------------------------------------------------------------------
extern "C" void kernel_launch(void* const* d_in, const int* in_sizes, int n_in,
                              void* d_out, int out_size, void* d_ws, size_t ws_size,
                              hipStream_t stream) {
    const float* coeff     = (const float*)d_in[0];
    const float* meanshape = (const float*)d_in[1];
    const float* idBase    = (const float*)d_in[2];
    const float* exBase    = (const float*)d_in[3];
    const float* meantex   = (const float*)d_in[4];
    const float* texBase   = (const float*)d_in[5];
    const int*   tri       = (const int*)d_in[6];
    const int*   point_buf = (const int*)d_in[7];
    const int*   kp        = (const int*)d_in[8];

    const int NV3 = in_sizes[1];
    const int NV  = NV3 / 3;
    const int NF  = in_sizes[6] / 3;
    const int NKP = in_sizes[8];
    const int Kid = in_sizes[2] / NV3;   // 80
    const int Kex = in_sizes[3] / NV3;   // 64
    const int Ktx = in_sizes[5] / NV3;   // 80

    float* out = (float*)d_out;
    float* fc    = out;                                         // face_color (B,NV,3)
    float* lmsO  = out + (size_t)B_ROWS * NV * 3;               // lms (B,68,2)
    float* shp   = lmsO + (size_t)B_ROWS * NKP * 2;             // shape_t (B,NV,3)

    float* ws        = (float*)d_ws;
    float* rot       = ws;                  // 128*9
    float* gamma_adj = ws + 1152;           // 128*27
    float* center    = ws + 1152 + 3456;    // 3
    float* fnbuf     = ws + 4640;           // B*(NF+1)*3

    precompute_kernel<<<1, 128, 0, stream>>>(coeff, meanshape, NV, rot, gamma_adj, center);

    const int gemmBlocks = (NV3 + 127) / 128;  // 8 waves/block, 16 cols/wave
    // shape (centered) -> shape_t slot
    gemm16_wmma<<<gemmBlocks, 256, 0, stream>>>(coeff, 0, idBase, Kid, exBase, Kex,
                                                meanshape, center, shp, NV3);
    // tex -> face_color slot
    gemm16_wmma<<<gemmBlocks, 256, 0, stream>>>(coeff, Kid + Kex, texBase, Ktx, nullptr, 0,
                                                meantex, nullptr, fc, NV3);

    const long long fnTot = (long long)B_ROWS * (NF + 1);
    face_normal_kernel<<<(int)((fnTot + 255) / 256), 256, 0, stream>>>(shp, tri, fnbuf, NV3, NF);

    const long long vTot = (long long)B_ROWS * NV;
    rigid_kernel<<<(int)((vTot + 255) / 256), 256, 0, stream>>>(shp, coeff, rot, NV);
    shade_kernel<<<(int)((vTot + 255) / 256), 256, 0, stream>>>(fc, fnbuf, point_buf, rot,
                                                                gamma_adj, NV, NF);
    lms_kernel<<<(B_ROWS * NKP + 255) / 256, 256, 0, stream>>>(shp, kp, lmsO, NV, NKP);
}